// CONV_GNN_28887950033711
// MI455X (gfx1250) — compile-verified
//
#include <hip/hip_runtime.h>
#include <hip/hip_bf16.h>

typedef __attribute__((ext_vector_type(2))) float v2f;
typedef __attribute__((ext_vector_type(8))) float v8f;

#define N_NODES 100000
#define N_EDGES 1600000
#define IN_F 128
#define HID 64

// ---------------------------------------------------------------------------
// Degree / normalization kernels (run once, reused by all 5 layers)
// ---------------------------------------------------------------------------
__global__ void gcn_deg_init(float* __restrict__ deg, int n) {
    int i = blockIdx.x * blockDim.x + threadIdx.x;
    if (i < n) deg[i] = 1.0f;   // self-loop contribution
}

__global__ void gcn_deg_acc(const int* __restrict__ col, float* __restrict__ deg, int nE) {
    int e = blockIdx.x * blockDim.x + threadIdx.x;
    if (e < nE) atomicAdd(&deg[col[e]], 1.0f);
}

__global__ void gcn_deg_rsqrt(float* __restrict__ deg, int n) {
    int i = blockIdx.x * blockDim.x + threadIdx.x;
    if (i < n) deg[i] = rsqrtf(deg[i]);   // deg >= 1 always (self-loops)
}

// norm[e] = dinv[row[e]] * dinv[col[e]]  (6.4MB, amortized over 5 layers)
__global__ void gcn_edge_norm(const int* __restrict__ row, const int* __restrict__ col,
                              const float* __restrict__ dinv, float* __restrict__ norm, int nE) {
    int e = blockIdx.x * blockDim.x + threadIdx.x;
    if (e < nE) norm[e] = dinv[row[e]] * dinv[col[e]];
}

// ---------------------------------------------------------------------------
// f32 WMMA GEMM: C[N x 64] = A[N x K] * B[K x 64]
// One wave32 computes one 16x16 tile of C using V_WMMA_F32_16X16X4_F32.
// N_NODES % 16 == 0 and HID == 64 -> grid covers exactly, no guards, EXEC full.
// ---------------------------------------------------------------------------
__global__ __launch_bounds__(256) void gcn_gemm_wmma(const float* __restrict__ A,
                                                     const float* __restrict__ B,
                                                     float* __restrict__ C, int K) {
    const int wave   = blockIdx.x * (blockDim.x >> 5) + (threadIdx.x >> 5);
    const int lane   = threadIdx.x & 31;
    const int tile_m = wave >> 2;      // node tile (16 rows)
    const int tile_n = wave & 3;       // output-feature tile (16 cols of 64)

    const int mrow  = tile_m * 16 + (lane & 15);   // A row for this lane
    const int ncol  = tile_n * 16 + (lane & 15);   // B col for this lane
    const int khalf = (lane >> 4) * 2;             // lanes 16-31 hold K+2,K+3

    v8f acc = {};
    for (int kk = 0; kk < K; kk += 4) {
        // A 16x4 fragment: lane holds A[mrow][kk+khalf], A[mrow][kk+khalf+1]
        v2f a = *(const v2f*)(A + (size_t)mrow * K + kk + khalf);
        // B 4x16 fragment: lane holds B[kk+khalf][ncol], B[kk+khalf+1][ncol]
        v2f b;
        b.x = B[(size_t)(kk + khalf) * HID + ncol];
        b.y = B[(size_t)(kk + khalf + 1) * HID + ncol];
        acc = __builtin_amdgcn_wmma_f32_16x16x4_f32(
            /*neg_a=*/false, a, /*neg_b=*/false, b,
            /*c_mod=*/(short)0, acc, /*reuse_a=*/false, /*reuse_b=*/false);
    }

    // D layout: VGPR v -> row (v | v+8), lanes 0-15 -> cols 0-15
    const int mbase = tile_m * 16 + ((lane >> 4) << 3);
    const int nout  = tile_n * 16 + (lane & 15);
#pragma unroll
    for (int v = 0; v < 8; ++v)
        C[(size_t)(mbase + v) * HID + nout] = acc[v];
}

// ---------------------------------------------------------------------------
// agg[i] = hw[i] * dinv[i]^2  (self-loop term; fully initializes agg).
// float4: tid -> node i = tid>>4, feature quad q = tid&15
// ---------------------------------------------------------------------------
__global__ void gcn_self_init(const float* __restrict__ hw, const float* __restrict__ dinv,
                              float* __restrict__ agg, int n) {
    int tid = blockIdx.x * blockDim.x + threadIdx.x;
    int i = tid >> 4;
    if (i >= n) return;
    float di = dinv[i];
    float s = di * di;
    float4 h4 = ((const float4*)hw)[tid];
    float4 o;
    o.x = h4.x * s; o.y = h4.y * s; o.z = h4.z * s; o.w = h4.w * s;
    ((float4*)agg)[tid] = o;
}

// ---------------------------------------------------------------------------
// Edge scatter: agg[col] += hw[row] * norm[e]
// 16 threads per edge; each thread: one b128 gather + 4 f32 atomics.
// All traffic L2-resident (working set ~84MB < 192MB global L2).
// ---------------------------------------------------------------------------
__global__ void gcn_edge_scatter(const float* __restrict__ hw, const float* __restrict__ norm,
                                 const int* __restrict__ row, const int* __restrict__ col,
                                 float* __restrict__ agg, int nE) {
    long long tid = (long long)blockIdx.x * blockDim.x + threadIdx.x;
    int e = (int)(tid >> 4);
    int q = (int)(tid & 15);            // feature quad: features 4q..4q+3
    if (e >= nE) return;
    int r = row[e];
    int c = col[e];
    float s = norm[e];
    float4 h4 = ((const float4*)(hw + (size_t)r * HID))[q];
    float* dst = agg + (size_t)c * HID + 4 * q;
    atomicAdd(dst + 0, h4.x * s);
    atomicAdd(dst + 1, h4.y * s);
    atomicAdd(dst + 2, h4.z * s);
    atomicAdd(dst + 3, h4.w * s);
}

// ---------------------------------------------------------------------------
// out = relu(agg + bias)   (float4)
// ---------------------------------------------------------------------------
__global__ void gcn_bias_relu(const float* __restrict__ agg, const float* __restrict__ bias,
                              float* __restrict__ out, int n) {
    int tid = blockIdx.x * blockDim.x + threadIdx.x;
    if (tid >= n * (HID / 4)) return;
    float4 a4 = ((const float4*)agg)[tid];
    float4 b4 = ((const float4*)bias)[tid & 15];
    float4 o;
    o.x = fmaxf(a4.x + b4.x, 0.0f);
    o.y = fmaxf(a4.y + b4.y, 0.0f);
    o.z = fmaxf(a4.z + b4.z, 0.0f);
    o.w = fmaxf(a4.w + b4.w, 0.0f);
    ((float4*)out)[tid] = o;
}

// ---------------------------------------------------------------------------
extern "C" void kernel_launch(void* const* d_in, const int* in_sizes, int n_in,
                              void* d_out, int out_size, void* d_ws, size_t ws_size,
                              hipStream_t stream) {
    (void)in_sizes; (void)n_in; (void)out_size; (void)ws_size;

    const float* x   = (const float*)d_in[0];
    const int*   ei  = (const int*)d_in[1];
    const int*   row = ei;              // edge_index[0]
    const int*   col = ei + N_EDGES;    // edge_index[1]

    const float* W[5] = { (const float*)d_in[2], (const float*)d_in[4], (const float*)d_in[6],
                          (const float*)d_in[8], (const float*)d_in[10] };
    const float* b[5] = { (const float*)d_in[3], (const float*)d_in[5], (const float*)d_in[7],
                          (const float*)d_in[9], (const float*)d_in[11] };

    float* ws   = (float*)d_ws;
    float* dinv = ws;                                   // N floats (pad to 100096)
    float* norm = ws + 100096;                          // E floats
    float* hw   = norm + N_EDGES;                       // N*64  (16B aligned)
    float* agg  = hw + (size_t)N_NODES * HID;           // N*64
    float* hbuf = agg + (size_t)N_NODES * HID;          // N*64

    const int TB = 256;
    const int gN    = (N_NODES + TB - 1) / TB;          // 391
    const int gE    = (N_EDGES + TB - 1) / TB;          // 6250
    const int gNF4  = (N_NODES * (HID / 4)) / TB;       // 6250
    const int gEF4  = (N_EDGES * 16) / TB;              // 100000
    const int gGemm = (N_NODES / 16) * 4 / 8;           // 25000 waves / 8 per block

    // normalization: deg -> dinv -> per-edge norm (deterministic every call)
    gcn_deg_init<<<gN, TB, 0, stream>>>(dinv, N_NODES);
    gcn_deg_acc<<<gE, TB, 0, stream>>>(col, dinv, N_EDGES);
    gcn_deg_rsqrt<<<gN, TB, 0, stream>>>(dinv, N_NODES);
    gcn_edge_norm<<<gE, TB, 0, stream>>>(row, col, dinv, norm, N_EDGES);

    const float* hin = x;
    int K = IN_F;
    for (int l = 0; l < 5; ++l) {
        gcn_gemm_wmma<<<gGemm, TB, 0, stream>>>(hin, W[l], hw, K);
        gcn_self_init<<<gNF4, TB, 0, stream>>>(hw, dinv, agg, N_NODES);
        gcn_edge_scatter<<<gEF4, TB, 0, stream>>>(hw, norm, row, col, agg, N_EDGES);
        float* out = (l == 4) ? (float*)d_out : hbuf;
        gcn_bias_relu<<<gNF4, TB, 0, stream>>>(agg, b[l], out, N_NODES);
        hin = out;
        K = HID;
    }
}